// GaussianLayer_24455543784029
// MI455X (gfx1250) — compile-verified
//
#include <hip/hip_runtime.h>
#include <hip/hip_bf16.h>
#include <math.h>

typedef __attribute__((ext_vector_type(2))) float v2f;
typedef __attribute__((ext_vector_type(8))) float v8f;

// out[p, k] = exp(-0.5*((x_p - m_k)/s_k)^2) / (sqrt(2pi)*s_k),  s_k = |stds_k| + 0.01
//
// Exponent evaluated as rank-3 GEMM on the matrix pipe:
//   arg(p,k) = x_p^2 * b0_k + x_p * b1_k + b2_k      (log2e folded into b*)
//   b0 = -0.5*log2e/s^2,  b1 = m*log2e/s^2,  b2 = -0.5*m^2*log2e/s^2
// via V_WMMA_F32_16X16X4_F32: A = 16 pixels x [x^2, x, 1, 0], B = 4 x 16 ks.
__global__ void __launch_bounds__(256)
gaussian_wmma_kernel(const float* __restrict__ x,
                     const float* __restrict__ means,
                     const float* __restrict__ stds,
                     float* __restrict__ out,
                     int P, int K) {
    const float LOG2E   = 1.44269504088896340736f;
    const float INV_PRE = 0.39894228040143267794f; // 1/sqrt(2*pi)

    const int lane = threadIdx.x & 31;
    const int lm   = lane & 15;            // N-index within 16-wide k tile / pixel row idx
    const int wave = threadIdx.x >> 5;
    const int pb   = (blockIdx.x * 8 + wave) * 16;   // 16 pixels per wave
    if (pb >= P) return;

    // ---- A matrix: 16x4 fp32, A[m] = [x^2, x, 1, 0] ----
    // Layout (ISA 7.12.2): lanes 0-15 VGPR0=K0, VGPR1=K1; lanes 16-31 VGPR0=K2, VGPR1=K3.
    const float xv = x[pb + lm];
    v2f a;
    if (lane < 16) { a.x = xv * xv; a.y = xv; }
    else           { a.x = 1.0f;    a.y = 0.0f; }

    const int rowoff = (lane < 16) ? 0 : 8;  // C layout: lanes 16-31 hold rows M=8..15
    const int ktiles = K >> 4;

    for (int kb = 0; kb < ktiles; ++kb) {
        const int k = kb * 16 + lm;

        // Per-k coefficients (means/stds are tiny and L1-resident).
        const float m    = means[k];
        const float s    = fabsf(stds[k]) + 0.01f;
        const float ivs  = 1.0f / s;
        const float iv2l = ivs * ivs * LOG2E;

        // ---- B matrix: 4x16 fp32 (rows K=0,2 in VGPR0 lane halves; K=1,3 in VGPR1) ----
        v2f b;
        b.x = (lane < 16) ? (-0.5f * iv2l)         // row 0: x^2 coeff
                          : (-0.5f * m * m * iv2l); // row 2: const term
        b.y = (lane < 16) ? (m * iv2l)             // row 1: x coeff
                          : 0.0f;                  // row 3: padding
        const float cn = ivs * INV_PRE;            // normalization for this k

        v8f acc = {0.f, 0.f, 0.f, 0.f, 0.f, 0.f, 0.f, 0.f};
        acc = __builtin_amdgcn_wmma_f32_16x16x4_f32(
                  /*neg_a=*/false, a, /*neg_b=*/false, b,
                  /*c_mod=*/(short)0, acc, /*reuse_a=*/false, /*reuse_b=*/false);

        // C layout: VGPR r holds (M = r + rowoff, N = lm). Apply exp2 + scale, NT store.
        // Output is [P, K] row-major: each half-wave writes 64B contiguous.
#pragma unroll
        for (int r = 0; r < 8; ++r) {
            const float v = __builtin_amdgcn_exp2f(acc[r]) * cn;
            const size_t o = (size_t)(pb + rowoff + r) * (size_t)K + (size_t)k;
            __builtin_nontemporal_store(v, &out[o]);   // 537MB stream > 192MB L2 -> NT
        }
    }
}

extern "C" void kernel_launch(void* const* d_in, const int* in_sizes, int n_in,
                              void* d_out, int out_size, void* d_ws, size_t ws_size,
                              hipStream_t stream) {
    const float* inp   = (const float*)d_in[0];   // [B, N, N] fp32, flat P elements
    const float* means = (const float*)d_in[1];   // [1, K]
    const float* stds  = (const float*)d_in[2];   // [1, K]
    float* out = (float*)d_out;                   // [B, N, N, K] fp32

    const int P = in_sizes[0];                    // 4*512*512 = 1,048,576
    const int K = in_sizes[1];                    // 128

    // 256 threads = 8 waves = 128 pixels per block
    const int blocks = (P + 127) / 128;
    gaussian_wmma_kernel<<<blocks, 256, 0, stream>>>(inp, means, stds, out, P, K);
}